// SPDNet_34754875359646
// MI455X (gfx1250) — compile-verified
//
#include <hip/hip_runtime.h>
#include <hip/hip_bf16.h>
#include <math.h>

typedef float v2f __attribute__((ext_vector_type(2)));
typedef float v8f __attribute__((ext_vector_type(8)));

#define EPSV 1e-4f

// ---------------------------------------------------------------------------
// fp32 WMMA helper: D = A(16x4) * B(4x16) + C, full fp32 precision.
// A frag: lane l<16 holds A[m0+l][k0+0], A[m0+l][k0+1]; lanes 16..31 hold K+2,K+3.
// B frag: lane l<16 holds B[k0+0][n0+l], B[k0+1][n0+l]; lanes 16..31 hold K+2,K+3.
// C/D frag: vgpr r: lanes 0-15 -> (m0+r, n0+l); lanes 16-31 -> (m0+8+r, n0+l).
// ---------------------------------------------------------------------------
__device__ __forceinline__ v8f wmma_f32_k4(v2f a, v2f b, v8f c) {
    return __builtin_amdgcn_wmma_f32_16x16x4_f32(false, a, false, b, (short)0, c,
                                                 false, false);
}

// ---------------------------------------------------------------------------
// Kernel 1: QR of W1 (64x48) and W2 (48x32) via modified Gram-Schmidt.
// Writes Wst1 = Q1^T (48x64) and Wst2 = Q2^T (32x48). Single block, 64 threads.
// ---------------------------------------------------------------------------
__device__ __forceinline__ float reduce64(float* red, int t, float v) {
    red[t] = v;
    __syncthreads();
    for (int off = 32; off > 0; off >>= 1) {
        if (t < off) red[t] += red[t + off];
        __syncthreads();
    }
    float r = red[0];
    __syncthreads();
    return r;
}

__global__ void qr_kernel(const float* __restrict__ W1, const float* __restrict__ W2,
                          float* __restrict__ wst1, float* __restrict__ wst2) {
    __shared__ float Q[64 * 48];
    __shared__ float red[64];
    const int t = threadIdx.x;  // 64 threads

    // ---- W1: 64 rows x 48 cols ----
    for (int idx = t; idx < 64 * 48; idx += 64) Q[idx] = W1[idx];
    __syncthreads();
    for (int j = 0; j < 48; ++j) {
        for (int k = 0; k < j; ++k) {
            float r = reduce64(red, t, Q[t * 48 + k] * Q[t * 48 + j]);
            Q[t * 48 + j] -= r * Q[t * 48 + k];
            __syncthreads();
        }
        float nrm = sqrtf(reduce64(red, t, Q[t * 48 + j] * Q[t * 48 + j]) + 1e-30f);
        Q[t * 48 + j] /= nrm;
        __syncthreads();
    }
    for (int idx = t; idx < 48 * 64; idx += 64) {
        int j = idx / 64, i = idx % 64;
        wst1[idx] = Q[i * 48 + j];
    }
    __syncthreads();

    // ---- W2: 48 rows x 32 cols ----
    for (int idx = t; idx < 48 * 32; idx += 64) Q[idx] = W2[idx];
    __syncthreads();
    for (int j = 0; j < 32; ++j) {
        for (int k = 0; k < j; ++k) {
            float prod = (t < 48) ? Q[t * 32 + k] * Q[t * 32 + j] : 0.f;
            float r = reduce64(red, t, prod);
            if (t < 48) Q[t * 32 + j] -= r * Q[t * 32 + k];
            __syncthreads();
        }
        float sq = (t < 48) ? Q[t * 32 + j] * Q[t * 32 + j] : 0.f;
        float nrm = sqrtf(reduce64(red, t, sq) + 1e-30f);
        if (t < 48) Q[t * 32 + j] /= nrm;
        __syncthreads();
    }
    for (int idx = t; idx < 32 * 48; idx += 64) {
        int j = idx / 48, i = idx % 48;
        wst2[idx] = Q[i * 32 + j];
    }
}

// ---------------------------------------------------------------------------
// Kernel 2: Y1[b] = Wst1 (48x64) * X[b] (64x64) * Wst1^T (64x48)
// One block (12 waves = 384 threads) per batch element. fp32 WMMA 16x16x4.
// ---------------------------------------------------------------------------
__global__ void bimap1_kernel(const float* __restrict__ X,
                              const float* __restrict__ wst1,
                              float* __restrict__ Y1) {
    __shared__ float Xs[64 * 66];  // padded stride 66 (bank-safe)
    __shared__ float Ts[48 * 68];  // padded stride 68 (bank-safe column reads)
    const int b = blockIdx.x;
    const float* Xb = X + (size_t)b * 4096;
    for (int t = threadIdx.x; t < 4096; t += blockDim.x)
        Xs[(t >> 6) * 66 + (t & 63)] = Xb[t];
    __syncthreads();

    const int wave = threadIdx.x >> 5;
    const int lane = threadIdx.x & 31;
    const int l = lane & 15;
    const int hi = lane >> 4;

    // Stage 1: T = Wst1 @ X  (48x64), tiles 3x4 -> 12 waves
    {
        const int m0 = (wave >> 2) * 16, n0 = (wave & 3) * 16;
        v8f acc = {0.f, 0.f, 0.f, 0.f, 0.f, 0.f, 0.f, 0.f};
        for (int k0 = 0; k0 < 64; k0 += 4) {
            v2f a, bf;
            a.x = wst1[(m0 + l) * 64 + k0 + 2 * hi];
            a.y = wst1[(m0 + l) * 64 + k0 + 2 * hi + 1];
            bf.x = Xs[(k0 + 2 * hi) * 66 + n0 + l];
            bf.y = Xs[(k0 + 2 * hi + 1) * 66 + n0 + l];
            acc = wmma_f32_k4(a, bf, acc);
        }
        for (int r = 0; r < 8; ++r)
            Ts[(m0 + r + 8 * hi) * 68 + n0 + l] = acc[r];
    }
    __syncthreads();

    // Stage 2: Y1 = T @ Wst1^T  (48x48), tiles 3x3 -> waves 0..8 (wave-uniform)
    if (wave < 9) {
        const int m0 = (wave / 3) * 16, n0 = (wave % 3) * 16;
        v8f acc = {0.f, 0.f, 0.f, 0.f, 0.f, 0.f, 0.f, 0.f};
        for (int k0 = 0; k0 < 64; k0 += 4) {
            v2f a, bf;
            a.x = Ts[(m0 + l) * 68 + k0 + 2 * hi];
            a.y = Ts[(m0 + l) * 68 + k0 + 2 * hi + 1];
            bf.x = wst1[(n0 + l) * 64 + k0 + 2 * hi];  // B[k][n] = Wst1[n][k]
            bf.y = wst1[(n0 + l) * 64 + k0 + 2 * hi + 1];
            acc = wmma_f32_k4(a, bf, acc);
        }
        float* Yb = Y1 + (size_t)b * 2304;
        for (int r = 0; r < 8; ++r)
            Yb[(m0 + r + 8 * hi) * 48 + n0 + l] = acc[r];
    }
}

// ---------------------------------------------------------------------------
// Kernel 3: Y2[b] = Wst2 (32x48) * Y1[b] (48x48) * Wst2^T (48x32)
// One block (6 waves = 192 threads) per batch element.
// ---------------------------------------------------------------------------
__global__ void bimap2_kernel(const float* __restrict__ Y1,
                              const float* __restrict__ wst2,
                              float* __restrict__ Y2) {
    __shared__ float Ys[48 * 50];
    __shared__ float Ts[32 * 52];
    const int b = blockIdx.x;
    const float* Yb = Y1 + (size_t)b * 2304;
    for (int t = threadIdx.x; t < 2304; t += blockDim.x)
        Ys[(t / 48) * 50 + (t % 48)] = Yb[t];
    __syncthreads();

    const int wave = threadIdx.x >> 5;
    const int lane = threadIdx.x & 31;
    const int l = lane & 15;
    const int hi = lane >> 4;

    // Stage 1: T = Wst2 @ Y1  (32x48), tiles 2x3 -> 6 waves
    {
        const int m0 = (wave / 3) * 16, n0 = (wave % 3) * 16;
        v8f acc = {0.f, 0.f, 0.f, 0.f, 0.f, 0.f, 0.f, 0.f};
        for (int k0 = 0; k0 < 48; k0 += 4) {
            v2f a, bf;
            a.x = wst2[(m0 + l) * 48 + k0 + 2 * hi];
            a.y = wst2[(m0 + l) * 48 + k0 + 2 * hi + 1];
            bf.x = Ys[(k0 + 2 * hi) * 50 + n0 + l];
            bf.y = Ys[(k0 + 2 * hi + 1) * 50 + n0 + l];
            acc = wmma_f32_k4(a, bf, acc);
        }
        for (int r = 0; r < 8; ++r)
            Ts[(m0 + r + 8 * hi) * 52 + n0 + l] = acc[r];
    }
    __syncthreads();

    // Stage 2: Y2 = T @ Wst2^T  (32x32), tiles 2x2 -> waves 0..3
    if (wave < 4) {
        const int m0 = (wave >> 1) * 16, n0 = (wave & 1) * 16;
        v8f acc = {0.f, 0.f, 0.f, 0.f, 0.f, 0.f, 0.f, 0.f};
        for (int k0 = 0; k0 < 48; k0 += 4) {
            v2f a, bf;
            a.x = Ts[(m0 + l) * 52 + k0 + 2 * hi];
            a.y = Ts[(m0 + l) * 52 + k0 + 2 * hi + 1];
            bf.x = wst2[(n0 + l) * 48 + k0 + 2 * hi];
            bf.y = wst2[(n0 + l) * 48 + k0 + 2 * hi + 1];
            acc = wmma_f32_k4(a, bf, acc);
        }
        float* Zb = Y2 + (size_t)b * 1024;
        for (int r = 0; r < 8; ++r)
            Zb[(m0 + r + 8 * hi) * 32 + n0 + l] = acc[r];
    }
}

// ---------------------------------------------------------------------------
// Kernel 4/5: batched cyclic Jacobi eigensolver (parallel round-robin ordering),
// in-place: M[b] <- V f(clamp(eig(M[b]+eps I))) V^T.
// DO_LOG=false: f(w)=clamp(w,eps,1e6)          (ReEig)
// DO_LOG=true:  f(w)=log(clamp(w,eps,1e6)+eps) (fused ReEig+LogEig)
// One 256-thread block per matrix; matrix + eigenvectors live in LDS.
// ---------------------------------------------------------------------------
template <int N, bool DO_LOG>
__global__ void jacobi_kernel(float* __restrict__ M) {
    constexpr int S = N + 1;   // padded stride -> conflict-free column access
    constexpr int NP = N / 2;  // independent rotations per round
    __shared__ float A[N * S];
    __shared__ float V[N * S];
    __shared__ float cs[NP], sn[NP];
    __shared__ int pp[NP], qq[NP];
    __shared__ float w[N];

    const int b = blockIdx.x;
    float* Mb = M + (size_t)b * N * N;
    const int tid = threadIdx.x;
    const int NT = blockDim.x;

    for (int t = tid; t < N * N; t += NT) {
        int i = t / N, j = t % N;
        A[i * S + j] = Mb[t] + ((i == j) ? EPSV : 0.f);
        V[i * S + j] = (i == j) ? 1.f : 0.f;
    }
    __syncthreads();

    for (int sweep = 0; sweep < 9; ++sweep) {
        for (int r = 0; r < N - 1; ++r) {
            if (tid < NP) {  // round-robin tournament pairing
                int k = tid;
                int p = (k == 0) ? 0 : (1 + (k - 1 + r) % (N - 1));
                int q = 1 + (N - 2 - k + r) % (N - 1);
                if (p > q) { int tmp = p; p = q; q = tmp; }
                float app = A[p * S + p], aqq = A[q * S + q], apq = A[p * S + q];
                float c = 1.f, s = 0.f;
                if (fabsf(apq) > 1e-12f) {
                    float th = 0.5f * (aqq - app) / apq;
                    float tt = copysignf(1.f, th) /
                               (fabsf(th) + sqrtf(1.f + th * th));
                    c = 1.f / sqrtf(1.f + tt * tt);
                    s = tt * c;
                }
                pp[k] = p; qq[k] = q; cs[k] = c; sn[k] = s;
            }
            __syncthreads();
            // rows: A <- J^T A (pairs disjoint -> fully parallel)
            for (int t = tid; t < NP * N; t += NT) {
                int k = t / N, j = t % N;
                int p = pp[k], q = qq[k];
                float c = cs[k], s = sn[k];
                float apj = A[p * S + j], aqj = A[q * S + j];
                A[p * S + j] = c * apj - s * aqj;
                A[q * S + j] = s * apj + c * aqj;
            }
            __syncthreads();
            // cols: A <- A J ; V <- V J
            for (int t = tid; t < NP * N; t += NT) {
                int k = t / N, i = t % N;
                int p = pp[k], q = qq[k];
                float c = cs[k], s = sn[k];
                float aip = A[i * S + p], aiq = A[i * S + q];
                A[i * S + p] = c * aip - s * aiq;
                A[i * S + q] = s * aip + c * aiq;
                float vip = V[i * S + p], viq = V[i * S + q];
                V[i * S + p] = c * vip - s * viq;
                V[i * S + q] = s * vip + c * viq;
            }
            __syncthreads();
        }
    }

    if (tid < N) {
        float wc = fminf(fmaxf(A[tid * S + tid], EPSV), 1e6f);
        w[tid] = DO_LOG ? logf(wc + EPSV) : wc;
    }
    __syncthreads();
    for (int t = tid; t < N * N; t += NT) {
        int i = t / N, j = t % N;
        float acc = 0.f;
        for (int k = 0; k < N; ++k)
            acc = fmaf(V[i * S + k] * w[k], V[j * S + k], acc);
        Mb[t] = acc;
    }
}

// ---------------------------------------------------------------------------
// Kernel 6: upper-tri vectorize + FC.  out[b][o] = fc_b[o] + sum_t vec[t]*fc_w[o][t]
// One thread per (b, o); fc_w (21 KB) and Y2 (34 MB) are L2-resident.
// ---------------------------------------------------------------------------
__global__ void fc_kernel(const float* __restrict__ Z, const float* __restrict__ fc_w,
                          const float* __restrict__ fc_b, float* __restrict__ out,
                          int B) {
    int gid = blockIdx.x * blockDim.x + threadIdx.x;
    if (gid >= B * 10) return;
    int b = gid / 10, o = gid % 10;
    const float* Zb = Z + (size_t)b * 1024;
    const float* wo = fc_w + o * 528;
    float acc = fc_b[o];
    int t = 0;
    for (int i = 0; i < 32; ++i)
        for (int j = i; j < 32; ++j, ++t)
            acc = fmaf(Zb[i * 32 + j], wo[t], acc);
    out[gid] = acc;
}

// ---------------------------------------------------------------------------
extern "C" void kernel_launch(void* const* d_in, const int* in_sizes, int n_in,
                              void* d_out, int out_size, void* d_ws, size_t ws_size,
                              hipStream_t stream) {
    const float* X = (const float*)d_in[0];      // (B, 64, 64)
    const float* W1 = (const float*)d_in[1];     // (64, 48)
    const float* W2 = (const float*)d_in[2];     // (48, 32)
    const float* fc_w = (const float*)d_in[3];   // (10, 528)
    const float* fc_b = (const float*)d_in[4];   // (10,)
    float* out = (float*)d_out;                  // (B, 10)

    const int B = in_sizes[0] / 4096;

    float* ws = (float*)d_ws;
    float* wst1 = ws;                              // 48*64
    float* wst2 = ws + 3072;                       // 32*48
    float* Y1 = ws + 4608;                         // B*48*48
    float* Y2 = Y1 + (size_t)B * 2304;             // B*32*32

    qr_kernel<<<1, 64, 0, stream>>>(W1, W2, wst1, wst2);
    bimap1_kernel<<<B, 384, 0, stream>>>(X, wst1, Y1);
    jacobi_kernel<48, false><<<B, 256, 0, stream>>>(Y1);
    bimap2_kernel<<<B, 192, 0, stream>>>(Y1, wst2, Y2);
    jacobi_kernel<32, true><<<B, 256, 0, stream>>>(Y2);
    fc_kernel<<<(B * 10 + 255) / 256, 256, 0, stream>>>(Y2, fc_w, fc_b, out, B);
}